// LDEPool1d_81527069213204
// MI455X (gfx1250) — compile-verified
//
#include <hip/hip_runtime.h>

typedef __attribute__((ext_vector_type(2))) float v2f;
typedef __attribute__((ext_vector_type(8))) float v8f;

#define Bsz 16
#define Tsz 800
#define Dsz 256
#define Ksz 64

#define TCH 32                 // timesteps per pool chunk
#define NCHUNK (Tsz / TCH)     // 25
#define XPAD 34                // LDS row stride (floats): distinct banks, 8B aligned

// explicit asynccnt waits (literal immediates)
#define WAIT_ASYNC_4() asm volatile("s_wait_asynccnt 0x4" ::: "memory")
#define WAIT_ASYNC_0() asm volatile("s_wait_asynccnt 0x0" ::: "memory")

// one 4-byte global -> LDS async copy (per-lane addresses), tracked by ASYNCcnt
__device__ __forceinline__ void async_g2l_b32(const float* g, unsigned lds_off) {
  asm volatile("global_load_async_to_lds_b32 %0, %1, off"
               :: "v"(lds_off), "v"(g) : "memory");
}

// ---------------------------------------------------------------------------
// prep: p2[k] = prec[k]^2 ; nmu[k] = ||mu_k||^2
// ---------------------------------------------------------------------------
__global__ __launch_bounds__(64) void lde_prep(const float* __restrict__ mu,
                                               const float* __restrict__ prec,
                                               float* __restrict__ p2,
                                               float* __restrict__ nmu) {
  int k = threadIdx.x;                       // 0..63
  float p = prec[k];
  p2[k] = p * p;
  const float* m = mu + (size_t)k * Dsz;
  float s = 0.f;
#pragma unroll 8
  for (int d = 0; d < Dsz; ++d) { float v = m[d]; s += v * v; }
  nmu[k] = s;
}

// ---------------------------------------------------------------------------
// logits+softmax: one wave per (b, 16-row time tile). WMMA computes
// dot(b,t,k)=x_bt . mu_k for 16 t-rows x 64 k in 4 f32 accumulators;
// softmax over k; writes r transposed as (B,K,T).
// ---------------------------------------------------------------------------
__global__ __launch_bounds__(32) void lde_logits(const float* __restrict__ x,
                                                 const float* __restrict__ mu,
                                                 const float* __restrict__ p2,
                                                 const float* __restrict__ nmu,
                                                 float* __restrict__ r_t) {
  __shared__ float lds[16 * 68];   // dot tile, padded stride
  __shared__ float sp2[Ksz];
  __shared__ float snm[Ksz];

  const int lane = threadIdx.x;
  const int lm   = lane & 15;
  const int half = lane >> 4;
  const int t0   = blockIdx.x * 16;   // 0..784
  const int b    = blockIdx.y;        // 0..15

  sp2[lane]      = p2[lane];
  sp2[lane + 32] = p2[lane + 32];
  snm[lane]      = nmu[lane];
  snm[lane + 32] = nmu[lane + 32];

  // A layout (16x4 f32): lanes 0-15 rows 0-15 depth {0,1}; lanes 16-31 depth {2,3}
  const float* xrow = x + ((size_t)(b * Tsz + t0 + lm)) * Dsz + half * 2;
  const float* mu0 = mu + (size_t)( 0 + lm) * Dsz + half * 2;
  const float* mu1 = mu + (size_t)(16 + lm) * Dsz + half * 2;
  const float* mu2 = mu + (size_t)(32 + lm) * Dsz + half * 2;
  const float* mu3 = mu + (size_t)(48 + lm) * Dsz + half * 2;

  v8f acc0 = {}, acc1 = {}, acc2 = {}, acc3 = {};
  float xsq = 0.f;

  for (int d0 = 0; d0 < Dsz; d0 += 4) {
    v2f a  = *(const v2f*)(xrow + d0);
    xsq += a.x * a.x + a.y * a.y;
    v2f b0 = *(const v2f*)(mu0 + d0);
    v2f b1 = *(const v2f*)(mu1 + d0);
    v2f b2 = *(const v2f*)(mu2 + d0);
    v2f b3 = *(const v2f*)(mu3 + d0);
    acc0 = __builtin_amdgcn_wmma_f32_16x16x4_f32(false, a, false, b0, (short)0, acc0, false, false);
    acc1 = __builtin_amdgcn_wmma_f32_16x16x4_f32(false, a, false, b1, (short)0, acc1, false, false);
    acc2 = __builtin_amdgcn_wmma_f32_16x16x4_f32(false, a, false, b2, (short)0, acc2, false, false);
    acc3 = __builtin_amdgcn_wmma_f32_16x16x4_f32(false, a, false, b3, (short)0, acc3, false, false);
  }

  // D layout: vgpr j, lanes0-15 -> (m=j, n=lane), lanes16-31 -> (m=j+8, n=lane-16)
#pragma unroll
  for (int j = 0; j < 8; ++j) {
    int m = j + half * 8;
    lds[m * 68 +  0 + lm] = acc0[j];
    lds[m * 68 + 16 + lm] = acc1[j];
    lds[m * 68 + 32 + lm] = acc2[j];
    lds[m * 68 + 48 + lm] = acc3[j];
  }
  __syncthreads();

  // both lane l and lane l+16 hold full ||x_row l||^2 after xor-16 combine
  float xn = xsq + __shfl_xor(xsq, 16, 32);

  const int m = lm;              // row owned by this lane
  const int kb = half * 32;      // k range owned by this lane

  float mx = -3.4e38f;
#pragma unroll 4
  for (int kk = 0; kk < 32; ++kk) {
    int k = kb + kk;
    float llk = -sp2[k] * (xn - 2.f * lds[m * 68 + k] + snm[k]);
    mx = fmaxf(mx, llk);
  }
  mx = fmaxf(mx, __shfl_xor(mx, 16, 32));   // full-row max

  float ssum = 0.f;
#pragma unroll 4
  for (int kk = 0; kk < 32; ++kk) {
    int k = kb + kk;
    float llk = -sp2[k] * (xn - 2.f * lds[m * 68 + k] + snm[k]);
    float e = __expf(llk - mx);
    lds[m * 68 + k] = e;                    // overwrite dot with exp
    ssum += e;
  }
  ssum += __shfl_xor(ssum, 16, 32);         // full-row sum
  float inv = 1.f / ssum;

#pragma unroll 4
  for (int kk = 0; kk < 32; ++kk) {
    int k = kb + kk;
    float r = lds[m * 68 + k] * inv;
    r_t[((size_t)(b * Ksz + k)) * Tsz + t0 + m] = r;   // (B,K,T) layout
  }
}

// ---------------------------------------------------------------------------
// colsum: S[b,k] = sum_t r_t[b,k,t]  (deterministic tree reduction)
// ---------------------------------------------------------------------------
__global__ __launch_bounds__(32) void lde_colsum(const float* __restrict__ r_t,
                                                 float* __restrict__ S) {
  int idx = blockIdx.x;                     // b*64+k, 0..1023
  const float* p = r_t + (size_t)idx * Tsz;
  int lane = threadIdx.x;
  float s = 0.f;
  for (int t = lane; t < Tsz; t += 32) s += p[t];
#pragma unroll
  for (int off = 16; off > 0; off >>= 1) s += __shfl_xor(s, off, 32);
  if (lane == 0) S[idx] = s;
}

// ---------------------------------------------------------------------------
// pool: 4-wave blocks; wave w owns k-tile w, all waves share a double-buffered
// x tile staged in LDS by global_load_async_to_lds_b32 (transposed [d][t]).
// acc(k,d) = sum_t r(b,t,k)*x(b,t,d) via f32 WMMA; epilogue (acc-S*mu)/(S+eps)
// ---------------------------------------------------------------------------
__global__ __launch_bounds__(128) void lde_pool(const float* __restrict__ x,
                                                const float* __restrict__ mu,
                                                const float* __restrict__ r_t,
                                                const float* __restrict__ S,
                                                float* __restrict__ out) {
  __shared__ __align__(16) float xb[2][16 * XPAD];  // [buf][d*XPAD + tloc]

  const int tid  = threadIdx.x;     // 0..127
  const int lane = tid & 31;
  const int wave = tid >> 5;        // 0..3 -> k-tile
  const int lm   = lane & 15;
  const int half = lane >> 4;
  const int d0 = blockIdx.x * 16;   // 0..240
  const int b  = blockIdx.y;        // 0..15
  const int k0 = wave * 16;

  // loader role: thread -> (d column, 4 consecutive timesteps)
  const int ld_d  = tid & 15;       // 0..15
  const int ld_t4 = (tid >> 4) * 4; // 0,4,..,28
  const float* gx = x + (size_t)(b * Tsz) * Dsz + d0 + ld_d;
  const unsigned lbase0 = (unsigned)(uintptr_t)&xb[0][ld_d * XPAD + ld_t4];
  const unsigned lbase1 = (unsigned)(uintptr_t)&xb[1][ld_d * XPAD + ld_t4];

  // A operand: r_t rows are contiguous in t thanks to (B,K,T) layout
  const float* ar = r_t + ((size_t)(b * Ksz + k0 + lm)) * Tsz + half * 2;

  v8f acc = {};

  // prologue: two chunks in flight
#pragma unroll
  for (int j = 0; j < 4; ++j)
    async_g2l_b32(gx + (size_t)(0 * TCH + ld_t4 + j) * Dsz, lbase0 + 4u * j);
#pragma unroll
  for (int j = 0; j < 4; ++j)
    async_g2l_b32(gx + (size_t)(1 * TCH + ld_t4 + j) * Dsz, lbase1 + 4u * j);

  for (int c = 0; c < NCHUNK; ++c) {
    if (c + 1 < NCHUNK) { WAIT_ASYNC_4(); } else { WAIT_ASYNC_0(); }
    __syncthreads();                          // chunk c visible to all waves

    const float* bbuf = &xb[c & 1][0];
    const float* a0   = ar + c * TCH;
#pragma unroll
    for (int s = 0; s < TCH / 4; ++s) {       // 8 WMMA steps per chunk
      v2f a  = *(const v2f*)(a0 + s * 4);
      v2f bb = *(const v2f*)(bbuf + lm * XPAD + s * 4 + half * 2);  // ds_load_b64
      acc = __builtin_amdgcn_wmma_f32_16x16x4_f32(false, a, false, bb,
                                                  (short)0, acc, false, false);
    }

    __syncthreads();                          // everyone done reading buf
    if (c + 2 < NCHUNK) {
      unsigned lb = (c & 1) ? lbase1 : lbase0;  // refill the buffer just freed
#pragma unroll
      for (int j = 0; j < 4; ++j)
        async_g2l_b32(gx + (size_t)((c + 2) * TCH + ld_t4 + j) * Dsz, lb + 4u * j);
    }
  }

#pragma unroll
  for (int j = 0; j < 8; ++j) {
    int m = j + half * 8;
    int k = k0 + m;
    float s   = S[b * Ksz + k];
    float muv = mu[(size_t)k * Dsz + d0 + lm];
    out[(size_t)b * (Ksz * Dsz) + (size_t)k * Dsz + d0 + lm] =
        (acc[j] - s * muv) / (s + 1e-9f);
  }
}

// ---------------------------------------------------------------------------
extern "C" void kernel_launch(void* const* d_in, const int* in_sizes, int n_in,
                              void* d_out, int out_size, void* d_ws, size_t ws_size,
                              hipStream_t stream) {
  const float* x    = (const float*)d_in[0];   // (16,800,256)
  const float* mu   = (const float*)d_in[1];   // (64,256)
  const float* prec = (const float*)d_in[2];   // (64,)
  float* out = (float*)d_out;                  // (16, 64*256)

  float* ws  = (float*)d_ws;
  float* p2  = ws;            // 64
  float* nmu = ws + 64;       // 64
  float* S   = ws + 128;      // 1024
  float* r_t = ws + 1152;     // 16*64*800 = 819200 floats (~3.3 MB)

  lde_prep  <<<1, 64, 0, stream>>>(mu, prec, p2, nmu);
  lde_logits<<<dim3(Tsz / 16, Bsz), 32, 0, stream>>>(x, mu, p2, nmu, r_t);
  lde_colsum<<<Bsz * Ksz, 32, 0, stream>>>(r_t, S);
  lde_pool  <<<dim3(Dsz / 16, Bsz), 128, 0, stream>>>(x, mu, r_t, S, out);
}